// SIMDIS_55525337202907
// MI455X (gfx1250) — compile-verified
//
#include <hip/hip_runtime.h>

typedef __attribute__((ext_vector_type(2))) float v2f;
typedef __attribute__((ext_vector_type(8))) float v8f;

#define NROWS 8192
#define DIM   512
#define TOPK  100
#define APITCH 513   // 513 % 64 == 1 -> 16-lane column reads hit distinct LDS banks

// ---------------- zero the S accumulator (must be re-zeroed every call) ----
__global__ __launch_bounds__(512) void k_zero(float* __restrict__ S) {
    S[threadIdx.x] = 0.0f;
}

// ---------------- per-row inverse norms: wave-per-row ----------------------
__global__ __launch_bounds__(256) void k_norms(const float* __restrict__ x,
                                               float* __restrict__ invn) {
    const int wave = threadIdx.x >> 5;
    const int lane = threadIdx.x & 31;
    const int row  = blockIdx.x * 8 + wave;
    const float4* x4 = (const float4*)(x + (size_t)row * DIM);
    float ss = 0.0f;
#pragma unroll
    for (int i = 0; i < 4; ++i) {
        float4 v = x4[lane + 32 * i];
        ss += v.x * v.x + v.y * v.y + v.z * v.z + v.w * v.w;
    }
#pragma unroll
    for (int off = 16; off > 0; off >>= 1) ss += __shfl_xor(ss, off, 32);
    if (lane == 0) invn[row] = 1.0f / fmaxf(sqrtf(ss), 1e-8f);
}

// ---------------- S[d] = sum_n x[n][d] * invn[n] ---------------------------
__global__ __launch_bounds__(256) void k_colsum(const float* __restrict__ x,
                                                const float* __restrict__ invn,
                                                float* __restrict__ S) {
    const int tid = threadIdx.x;           // owns columns 2*tid, 2*tid+1
    const int n0  = blockIdx.x * 64;
    const float2* x2 = (const float2*)x;
    float ax = 0.0f, ay = 0.0f;
    for (int r = 0; r < 64; ++r) {
        const int n = n0 + r;
        const float inv = invn[n];
        float2 v = x2[(size_t)n * 256 + tid];
        ax += v.x * inv;
        ay += v.y * inv;
    }
    atomicAdd(&S[2 * tid + 0], ax);
    atomicAdd(&S[2 * tid + 1], ay);
}

// ---------------- GEMV via V_WMMA_F32_16X16X4_F32 --------------------------
// One wave per 16-row tile. A = normalized rows (16 x 512) staged in LDS;
// B's 16 columns all equal S, so D[m][*] = u_m . S in every column.
__global__ __launch_bounds__(32) void k_gemv_wmma(const float* __restrict__ x,
                                                  const float* __restrict__ invn,
                                                  const float* __restrict__ S,
                                                  float* __restrict__ mean_sim) {
    __shared__ float a[16 * APITCH];
    __shared__ float s[DIM];
    const int lane = threadIdx.x;
    const int row0 = blockIdx.x * 16;

    // stage normalized A tile (coalesced float4 global loads)
    for (int r = 0; r < 16; ++r) {
        const float inv = invn[row0 + r];
        const float4* src = (const float4*)(x + (size_t)(row0 + r) * DIM);
#pragma unroll
        for (int i = 0; i < 4; ++i) {
            float4 v = src[lane + 32 * i];
            const int c = (lane + 32 * i) * 4;
            a[r * APITCH + c + 0] = v.x * inv;
            a[r * APITCH + c + 1] = v.y * inv;
            a[r * APITCH + c + 2] = v.z * inv;
            a[r * APITCH + c + 3] = v.w * inv;
        }
    }
    // stage S
#pragma unroll
    for (int i = lane; i < DIM / 4; i += 32) {
        float4 v = ((const float4*)S)[i];
        s[4 * i + 0] = v.x; s[4 * i + 1] = v.y;
        s[4 * i + 2] = v.z; s[4 * i + 3] = v.w;
    }
    __syncthreads();

    // ISA 16x4 f32 A layout: lanes 0-15 -> K={0,1}, lanes 16-31 -> K={2,3}
    const int m  = lane & 15;
    const int h2 = (lane >> 4) * 2;
    v8f c = {};
    for (int kb = 0; kb < DIM; kb += 4) {
        v2f av, bv;
        av.x = a[m * APITCH + kb + h2];
        av.y = a[m * APITCH + kb + h2 + 1];
        bv.x = s[kb + h2];
        bv.y = s[kb + h2 + 1];
        c = __builtin_amdgcn_wmma_f32_16x16x4_f32(false, av, false, bv,
                                                  (short)0, c, false, false);
    }

    // C/D layout: VGPR j = row j (lanes 0-15) / row j+8 (lanes 16-31); all
    // columns identical, so lane 0 and lane 16 each publish 8 row results.
    if ((lane & 15) == 0) {
        const int base = row0 + ((lane >> 4) ? 8 : 0);
#pragma unroll
        for (int j = 0; j < 8; ++j)
            mean_sim[base + j] = c[j] * (1.0f / (float)NROWS);
    }
}

// ---------------- rank-by-counting selection of the 100 smallest -----------
// (v[m]<v[n]) || (v[m]==v[n] && m<n) is a total order -> ranks 0..99 each
// written exactly once, matching stable ascending argsort semantics.
// Inner loop index is wave-uniform -> ds_load_b128 broadcast, conflict-free.
__global__ __launch_bounds__(256) void k_rank(const float* __restrict__ mean_sim,
                                              int* __restrict__ sel,
                                              float* __restrict__ out_idx) {
    __shared__ float v[NROWS];
    const int tid = threadIdx.x;
    for (int i = tid; i < NROWS; i += 256) v[i] = mean_sim[i];
    __syncthreads();
    const int n   = blockIdx.x * 256 + tid;
    const float vn = v[n];
    const float4* v4 = (const float4*)v;
    int rank = 0;
#pragma unroll 4
    for (int q = 0; q < NROWS / 4; ++q) {
        const float4 w = v4[q];
        const int m = 4 * q;
        rank += (w.x < vn) || (w.x == vn && (m + 0) < n);
        rank += (w.y < vn) || (w.y == vn && (m + 1) < n);
        rank += (w.z < vn) || (w.z == vn && (m + 2) < n);
        rank += (w.w < vn) || (w.w == vn && (m + 3) < n);
    }
    if (rank < TOPK) {
        sel[rank]     = n;
        out_idx[rank] = (float)n;
    }
}

// ---------------- gather the selected rows ---------------------------------
__global__ __launch_bounds__(256) void k_gather(const float* __restrict__ x,
                                                const int* __restrict__ sel,
                                                float* __restrict__ out) {
    const int b = blockIdx.x;                  // 0..99
    const int r = sel[b];
    for (int i = threadIdx.x; i < DIM; i += 256)
        out[(size_t)b * DIM + i] = x[(size_t)r * DIM + i];
}

extern "C" void kernel_launch(void* const* d_in, const int* in_sizes, int n_in,
                              void* d_out, int out_size, void* d_ws, size_t ws_size,
                              hipStream_t stream) {
    const float* x = (const float*)d_in[0];
    float* out = (float*)d_out;
    float* ws  = (float*)d_ws;

    // ws layout (floats): [0,8192) inv_norm | [8192,8704) S |
    //                     [8704,16896) mean_sim | [16896,+100) sel (int)
    float* invn = ws;
    float* S    = ws + NROWS;
    float* mean = ws + NROWS + DIM;
    int*   sel  = (int*)(ws + NROWS + DIM + NROWS);

    k_zero     <<<1,            512, 0, stream>>>(S);
    k_norms    <<<NROWS / 8,    256, 0, stream>>>(x, invn);
    k_colsum   <<<NROWS / 64,   256, 0, stream>>>(x, invn, S);
    k_gemv_wmma<<<NROWS / 16,    32, 0, stream>>>(x, invn, S, mean);
    k_rank     <<<NROWS / 256,  256, 0, stream>>>(mean, sel, out + TOPK * DIM);
    k_gather   <<<TOPK,         256, 0, stream>>>(x, sel, out);
}